// MAGNN_Agg_9560597201174
// MI455X (gfx1250) — compile-verified
//
#include <hip/hip_runtime.h>
#include <hip/hip_bf16.h>

#define DFEAT 128

typedef float v2f __attribute__((ext_vector_type(2)));
typedef float v8f __attribute__((ext_vector_type(8)));

// ---------------------------------------------------------------- zero
__global__ void zero_kernel(float* __restrict__ p, long n) {
    long i = (long)blockIdx.x * blockDim.x + threadIdx.x;
    long stride = (long)gridDim.x * blockDim.x;
    for (; i < n; i += stride) p[i] = 0.0f;
}

// ---------------------------------------------------------------- edge scatter
// One wave (32 lanes) per edge: lane l moves float4 at feature offset l*4.
//   acc[sidx[e]] += feat[gidx[e]] * (w ? w[e] : 1);  cnt[sidx[e]] += 1
__global__ void scatter_add_kernel(const float* __restrict__ feat,
                                   const int* __restrict__ gidx,
                                   const int* __restrict__ sidx,
                                   const float* __restrict__ w,   // nullable
                                   float* __restrict__ acc,
                                   float* __restrict__ cnt,
                                   int E) {
    long t = (long)blockIdx.x * blockDim.x + threadIdx.x;
    int e = (int)(t >> 5);
    int lane = (int)(t & 31);
    if (e >= E) return;
    int g = gidx[e];
    int s = sidx[e];
    float wt = (w != nullptr) ? w[e] : 1.0f;
    const float4* src = (const float4*)(feat + (long)g * DFEAT);
    float4 v = src[lane];
    v.x *= wt; v.y *= wt; v.z *= wt; v.w *= wt;
    float* dst = acc + (long)s * DFEAT + lane * 4;
    atomicAdd(dst + 0, v.x);
    atomicAdd(dst + 1, v.y);
    atomicAdd(dst + 2, v.z);
    atomicAdd(dst + 3, v.w);
    if (lane == 0) atomicAdd(cnt + s, 1.0f);
}

// ---------------------------------------------------------------- net = (msg/max(c,1) + x) * 0.5   (in place over msg)
__global__ void combine_kernel(float* __restrict__ msg,
                               const float* __restrict__ cnt,
                               const float* __restrict__ x, long n) {
    long i = (long)blockIdx.x * blockDim.x + threadIdx.x;
    if (i >= n) return;
    float c = fmaxf(cnt[i >> 7], 1.0f);   // row = i / 128
    msg[i] = (msg[i] / c + x[i]) * 0.5f;
}

// ---------------------------------------------------------------- acc /= max(cnt,1)  (in place)
__global__ void mean_kernel(float* __restrict__ acc,
                            const float* __restrict__ cnt, long n) {
    long i = (long)blockIdx.x * blockDim.x + threadIdx.x;
    if (i >= n) return;
    acc[i] = acc[i] / fmaxf(cnt[i >> 7], 1.0f);
}

// ---------------------------------------------------------------- Y = relu(A @ W^T + b), A:[M,128], W:[128,128]
// 8 waves/block; wave w owns output tile rows [m0,m0+16) x cols [16w,16w+16).
// f32 WMMA 16x16x4, K-loop of 32 chained accumulations. EXEC all-ones (M % 16 == 0).
__global__ __launch_bounds__(256)
void gemm_bias_relu_kernel(const float* __restrict__ A,
                           const float* __restrict__ W,
                           const float* __restrict__ bias,
                           float* __restrict__ Y) {
    const int wave = threadIdx.x >> 5;      // 0..7
    const int lane = threadIdx.x & 31;
    const int half = lane >> 4;             // 0: lanes 0-15, 1: lanes 16-31
    const int l16  = lane & 15;
    const int m0 = blockIdx.x * 16;
    const int n0 = wave * 16;

    // A 16x4 layout: lane<16 -> row=l16, K=k+{0,1}; lane>=16 -> row=l16, K=k+{2,3}
    // B 4x16 layout (B = W^T): vgpr v, lanes<16 -> K=k+v; lanes>=16 -> K=k+2+v; col=l16
    const float* arow = A + (long)(m0 + l16) * DFEAT;
    const float* brow = W + (long)(n0 + l16) * DFEAT;   // B[k][j] = W[j][k]

    v8f c = {0.f, 0.f, 0.f, 0.f, 0.f, 0.f, 0.f, 0.f};
#pragma unroll
    for (int k = 0; k < DFEAT; k += 4) {
        int ka = k + half * 2;
        v2f a = *(const v2f*)(arow + ka);
        v2f b = *(const v2f*)(brow + ka);
        c = __builtin_amdgcn_wmma_f32_16x16x4_f32(
            /*neg_a=*/false, a, /*neg_b=*/false, b,
            /*c_mod=*/(short)0, c, /*reuse_a=*/false, /*reuse_b=*/false);
    }

    // C/D layout: vgpr v -> row m0 + v + 8*half, col n0 + l16
    float bn = bias[n0 + l16];
#pragma unroll
    for (int v = 0; v < 8; ++v) {
        int row = m0 + v + half * 8;
        float val = c[v] + bn;
        Y[(long)row * DFEAT + (n0 + l16)] = fmaxf(val, 0.0f);
    }
}

// ---------------------------------------------------------------- attention combine
// One wave per node; lane handles 4 features; wave32 shuffle reduction for scores.
__global__ void attn_kernel(const float* __restrict__ y1,
                            const float* __restrict__ y2,
                            const float* __restrict__ y12,
                            const float* __restrict__ av,   // [3,128]
                            float* __restrict__ out, int N) {
    long t = (long)blockIdx.x * blockDim.x + threadIdx.x;
    int n = (int)(t >> 5);
    int lane = (int)(t & 31);
    if (n >= N) return;
    long base = (long)n * DFEAT + lane * 4;
    float4 a = *(const float4*)(y1 + base);
    float4 b = *(const float4*)(y2 + base);
    float4 c = *(const float4*)(y12 + base);
    float4 v0 = *(const float4*)(av + lane * 4);
    float4 v1 = *(const float4*)(av + DFEAT + lane * 4);
    float4 v2 = *(const float4*)(av + 2 * DFEAT + lane * 4);

    float s0 = a.x * v0.x + a.y * v0.y + a.z * v0.z + a.w * v0.w;
    float s1 = b.x * v1.x + b.y * v1.y + b.z * v1.z + b.w * v1.w;
    float s2 = c.x * v2.x + c.y * v2.y + c.z * v2.z + c.w * v2.w;
#pragma unroll
    for (int m = 16; m >= 1; m >>= 1) {
        s0 += __shfl_xor(s0, m, 32);
        s1 += __shfl_xor(s1, m, 32);
        s2 += __shfl_xor(s2, m, 32);
    }
    float mx = fmaxf(s0, fmaxf(s1, s2));
    float e0 = __expf(s0 - mx), e1 = __expf(s1 - mx), e2 = __expf(s2 - mx);
    float inv = 1.0f / (e0 + e1 + e2);
    e0 *= inv; e1 *= inv; e2 *= inv;

    float4 o;
    o.x = a.x * e0 + b.x * e1 + c.x * e2;
    o.y = a.y * e0 + b.y * e1 + c.y * e2;
    o.z = a.z * e0 + b.z * e1 + c.z * e2;
    o.w = a.w * e0 + b.w * e1 + c.w * e2;
    *(float4*)(out + base) = o;
}

// ---------------------------------------------------------------- host launch
extern "C" void kernel_launch(void* const* d_in, const int* in_sizes, int n_in,
                              void* d_out, int out_size, void* d_ws, size_t ws_size,
                              hipStream_t stream) {
    const float* x_node  = (const float*)d_in[0];
    const float* x1      = (const float*)d_in[1];
    const float* x2      = (const float*)d_in[2];
    const int* ei1_src   = (const int*)d_in[3];
    const int* ei1_dst   = (const int*)d_in[4];
    const int* ei2_src   = (const int*)d_in[5];
    const int* ei2_dst   = (const int*)d_in[6];
    const int* ei12_src  = (const int*)d_in[7];
    const int* ei12_dst  = (const int*)d_in[8];
    const float* ew1     = (const float*)d_in[9];
    const float* ew2     = (const float*)d_in[10];
    const float* W1      = (const float*)d_in[11];
    const float* b1      = (const float*)d_in[12];
    const float* W2      = (const float*)d_in[13];
    const float* b2      = (const float*)d_in[14];
    const float* W12     = (const float*)d_in[15];
    const float* b12     = (const float*)d_in[16];
    const float* att     = (const float*)d_in[17];

    const int N0  = in_sizes[0] / DFEAT;
    const int N1  = in_sizes[1] / DFEAT;
    const int N2  = in_sizes[2] / DFEAT;
    const int E1  = in_sizes[3];
    const int E2  = in_sizes[5];
    const int E12 = in_sizes[7];

    // ---- workspace partition (all accumulators contiguous -> single zero pass)
    float* ws = (float*)d_ws;
    long off = 0;
    float* msg1  = ws + off; off += (long)N1 * DFEAT;   // becomes net1
    float* msg2  = ws + off; off += (long)N2 * DFEAT;   // becomes net2
    float* msg12 = ws + off; off += (long)N2 * DFEAT;   // becomes net2b
    float* acc1  = ws + off; off += (long)N0 * DFEAT;
    float* acc2  = ws + off; off += (long)N0 * DFEAT;
    float* acc12 = ws + off; off += (long)N0 * DFEAT;
    float* cnt1  = ws + off; off += N1;
    float* cnt2  = ws + off; off += N2;
    float* cnt12 = ws + off; off += N2;
    float* cA1   = ws + off; off += N0;
    float* cA2   = ws + off; off += N0;
    float* cA12  = ws + off; off += N0;
    long zero_n = off;                                   // everything above needs zeroing
    float* y1  = ws + off; off += (long)N0 * DFEAT;      // fully overwritten, no zero
    float* y2  = ws + off; off += (long)N0 * DFEAT;
    float* y12 = ws + off; off += (long)N0 * DFEAT;

    const int TB = 256;

    // 0) zero accumulators + counts
    {
        long blocks = (zero_n + TB - 1) / TB;
        if (blocks > 262144) blocks = 262144;
        zero_kernel<<<(int)blocks, TB, 0, stream>>>(ws, zero_n);
    }

    // 1) forward scatters: msg1 <- x_node[ei1_src]*ew1 @ ei1_dst ; msg2 likewise
    {
        long tot = (long)E1 * 32;
        scatter_add_kernel<<<(int)((tot + TB - 1) / TB), TB, 0, stream>>>(
            x_node, ei1_src, ei1_dst, ew1, msg1, cnt1, E1);
    }
    {
        long tot = (long)E2 * 32;
        scatter_add_kernel<<<(int)((tot + TB - 1) / TB), TB, 0, stream>>>(
            x_node, ei2_src, ei2_dst, ew2, msg2, cnt2, E2);
    }

    // 2) net1 = (msg1/c + x1)/2 ; net2 = (msg2/c + x2)/2
    {
        long n = (long)N1 * DFEAT;
        combine_kernel<<<(int)((n + TB - 1) / TB), TB, 0, stream>>>(msg1, cnt1, x1, n);
    }
    {
        long n = (long)N2 * DFEAT;
        combine_kernel<<<(int)((n + TB - 1) / TB), TB, 0, stream>>>(msg2, cnt2, x2, n);
    }

    // 3) backward scatters to node accumulators
    {
        long tot = (long)E1 * 32;   // acc1[ei1_src] += net1[ei1_dst]
        scatter_add_kernel<<<(int)((tot + TB - 1) / TB), TB, 0, stream>>>(
            msg1, ei1_dst, ei1_src, nullptr, acc1, cA1, E1);
    }
    {
        long tot = (long)E2 * 32;   // acc2[ei2_src] += net2[ei2_dst]
        scatter_add_kernel<<<(int)((tot + TB - 1) / TB), TB, 0, stream>>>(
            msg2, ei2_dst, ei2_src, nullptr, acc2, cA2, E2);
    }

    // 4) metapath 1-2: msg12[ei12_dst] += net1[ei12_src]
    {
        long tot = (long)E12 * 32;
        scatter_add_kernel<<<(int)((tot + TB - 1) / TB), TB, 0, stream>>>(
            msg1, ei12_src, ei12_dst, nullptr, msg12, cnt12, E12);
    }
    {   // net2b = (msg12/c + x2)/2
        long n = (long)N2 * DFEAT;
        combine_kernel<<<(int)((n + TB - 1) / TB), TB, 0, stream>>>(msg12, cnt12, x2, n);
    }
    {   // acc12[ei2_src] += net2b[ei2_dst] * ew2
        long tot = (long)E2 * 32;
        scatter_add_kernel<<<(int)((tot + TB - 1) / TB), TB, 0, stream>>>(
            msg12, ei2_dst, ei2_src, ew2, acc12, cA12, E2);
    }

    // 5) means
    {
        long n = (long)N0 * DFEAT;
        int blocks = (int)((n + TB - 1) / TB);
        mean_kernel<<<blocks, TB, 0, stream>>>(acc1,  cA1,  n);
        mean_kernel<<<blocks, TB, 0, stream>>>(acc2,  cA2,  n);
        mean_kernel<<<blocks, TB, 0, stream>>>(acc12, cA12, n);
    }

    // 6) three WMMA GEMMs: y = relu(acc @ W^T + b)   (N0 % 16 == 0)
    {
        int blocks = N0 / 16;
        gemm_bias_relu_kernel<<<blocks, 256, 0, stream>>>(acc1,  W1,  b1,  y1);
        gemm_bias_relu_kernel<<<blocks, 256, 0, stream>>>(acc2,  W2,  b2,  y2);
        gemm_bias_relu_kernel<<<blocks, 256, 0, stream>>>(acc12, W12, b12, y12);
    }

    // 7) attention combine -> d_out
    {
        long tot = (long)N0 * 32;
        attn_kernel<<<(int)((tot + TB - 1) / TB), TB, 0, stream>>>(
            y1, y2, y12, att, (float*)d_out, N0);
    }
}